// SSCA_83889301226046
// MI455X (gfx1250) — compile-verified
//
#include <hip/hip_runtime.h>

typedef __attribute__((ext_vector_type(2))) float v2f;
typedef __attribute__((ext_vector_type(4))) float v4f;
typedef __attribute__((ext_vector_type(8))) float v8f;

#define BATCH 4
#define CCH   1024
#define NPIX  128          // H*W = 16*8
#define NATTR 21
#define KDIM  (NATTR*CCH)  // 21504

// ---------------------------------------------------------------------------
// Kernel 1: lq[b][o] = sum_k Qflat[b][k] * W3[o][k]   (M=4 GEMV, HBM-bound)
// One wave per W3 row; coalesced float4 streaming; Q rows live in L2/WGP$.
// ---------------------------------------------------------------------------
__global__ __launch_bounds__(256) void ssca_gemv_w3(
    const float* __restrict__ Q, const float* __restrict__ W3,
    float* __restrict__ lq) {
  const int wave = threadIdx.x >> 5;
  const int lane = threadIdx.x & 31;
  const int row  = blockIdx.x * 8 + wave;          // 2688 blocks * 8 waves = 21504
  const float* __restrict__ w = W3 + (size_t)row * KDIM;

  float a0 = 0.f, a1 = 0.f, a2 = 0.f, a3 = 0.f;
  for (int k = lane * 4; k < KDIM; k += 128) {     // 168 iterations, b128 loads
    __builtin_prefetch(w + k + 512, 0, 1);         // global_prefetch_b8 ahead of stream
    v4f wv = *(const v4f*)(w + k);
    v4f q0 = *(const v4f*)(Q + 0 * KDIM + k);
    v4f q1 = *(const v4f*)(Q + 1 * KDIM + k);
    v4f q2 = *(const v4f*)(Q + 2 * KDIM + k);
    v4f q3 = *(const v4f*)(Q + 3 * KDIM + k);
    a0 += wv.x * q0.x + wv.y * q0.y + wv.z * q0.z + wv.w * q0.w;
    a1 += wv.x * q1.x + wv.y * q1.y + wv.z * q1.z + wv.w * q1.w;
    a2 += wv.x * q2.x + wv.y * q2.y + wv.z * q2.z + wv.w * q2.w;
    a3 += wv.x * q3.x + wv.y * q3.y + wv.z * q3.z + wv.w * q3.w;
  }
  #pragma unroll
  for (int off = 16; off > 0; off >>= 1) {
    a0 += __shfl_down(a0, off, 32);
    a1 += __shfl_down(a1, off, 32);
    a2 += __shfl_down(a2, off, 32);
    a3 += __shfl_down(a3, off, 32);
  }
  if (lane == 0) {
    lq[0 * KDIM + row] = a0;
    lq[1 * KDIM + row] = a1;
    lq[2 * KDIM + row] = a2;
    lq[3 * KDIM + row] = a3;
  }
}

// ---------------------------------------------------------------------------
// Kernel 2: lf1 = W1 @ f + b1 ; lf2 = W2 @ f + b2   (per batch, f32 WMMA)
// M=1024(out-ch) x N=128(pixels) x K=1024(in-ch); 16x16 tile per wave,
// V_WMMA_F32_16X16X4_F32 K-loop, shared B fragment for both GEMMs.
// ---------------------------------------------------------------------------
__global__ __launch_bounds__(256) void ssca_lf_gemm(
    const float* __restrict__ feature,
    const float* __restrict__ W1, const float* __restrict__ b1,
    const float* __restrict__ W2, const float* __restrict__ b2,
    float* __restrict__ lf1, float* __restrict__ lf2) {
  const int waveId = blockIdx.x * 8 + (threadIdx.x >> 5); // 2048 waves total
  const int b   = waveId >> 9;                            // 512 waves per batch
  const int rem = waveId & 511;
  const int o0  = (rem >> 3) * 16;                        // 64 M-tiles
  const int n0  = (rem & 7) * 16;                         // 8 N-tiles
  const int lane = threadIdx.x & 31;
  const int m    = lane & 15;
  const int half = lane >> 4;                             // K half-select per ISA layout

  const float* __restrict__ f  = feature + (size_t)b * CCH * NPIX;
  const float* __restrict__ w1 = W1 + (size_t)(o0 + m) * CCH;
  const float* __restrict__ w2 = W2 + (size_t)(o0 + m) * CCH;

  v8f c1 = {};
  v8f c2 = {};
  for (int k = 0; k < CCH; k += 4) {
    const int kk = k + 2 * half;
    // A fragments (16x4 f32): lane holds row (o0+m), K pair {kk, kk+1}
    v2f af1 = *(const v2f*)(w1 + kk);
    v2f af2 = *(const v2f*)(w2 + kk);
    // B fragment (4x16 f32): lane holds col (n0+m), K pair {kk, kk+1}
    v2f bf;
    bf.x = f[(size_t)kk * NPIX + n0 + m];
    bf.y = f[(size_t)(kk + 1) * NPIX + n0 + m];
    c1 = __builtin_amdgcn_wmma_f32_16x16x4_f32(false, af1, false, bf,
                                               (short)0, c1, false, false);
    c2 = __builtin_amdgcn_wmma_f32_16x16x4_f32(false, af2, false, bf,
                                               (short)0, c2, false, false);
  }

  float* __restrict__ o1 = lf1 + (size_t)b * CCH * NPIX;
  float* __restrict__ o2 = lf2 + (size_t)b * CCH * NPIX;
  #pragma unroll
  for (int v = 0; v < 8; ++v) {
    const int row = o0 + v + 8 * half;   // C/D layout: lanes 16-31 hold M+8
    const int col = n0 + m;
    o1[(size_t)row * NPIX + col] = c1[v] + b1[row];
    o2[(size_t)row * NPIX + col] = c2[v] + b2[row];
  }
}

// ---------------------------------------------------------------------------
// Kernel 3: fused attention + output head, one block per batch.
//   g[n]   = sum_c fc_w[c] * lf2[c][n]
//   QK[i,n]= (1/32) * sum_c lq[c*21+i] * lf1[c][n]
//   A      = softmax over i
//   out[i] = sum_n A[i,n] * g[n] + fc_b        (ff intermediate eliminated)
// ---------------------------------------------------------------------------
__global__ __launch_bounds__(256) void ssca_attn(
    const float* __restrict__ lq, const float* __restrict__ lf1,
    const float* __restrict__ lf2, const float* __restrict__ fc_w,
    const float* __restrict__ fc_b, float* __restrict__ out) {
  const int b = blockIdx.x;
  const int t = threadIdx.x;
  __shared__ float gpart[256];
  __shared__ float g[NPIX];
  __shared__ float qk[NATTR * NPIX];

  const float* __restrict__ L1  = lf1 + (size_t)b * CCH * NPIX;
  const float* __restrict__ L2  = lf2 + (size_t)b * CCH * NPIX;
  const float* __restrict__ lqb = lq + (size_t)b * KDIM;

  // g[n] = fc_w . lf2[:,n]   (split the c-sum over two thread groups)
  {
    const int n  = t & 127;
    const int c0 = (t >> 7) * 512;
    float s = 0.f;
    for (int c = 0; c < 512; ++c)
      s += fc_w[c0 + c] * L2[(size_t)(c0 + c) * NPIX + n];
    gpart[t] = s;
  }
  __syncthreads();
  if (t < NPIX) g[t] = gpart[t] + gpart[t + 128];
  __syncthreads();

  // QK[i][n], scaled by 1/sqrt(C) = 1/32
  for (int idx = t; idx < NATTR * NPIX; idx += 256) {
    const int i = idx >> 7;
    const int n = idx & 127;
    float s = 0.f;
    for (int c = 0; c < CCH; ++c)
      s += lqb[c * NATTR + i] * L1[(size_t)c * NPIX + n];
    qk[idx] = s * 0.03125f;
  }
  __syncthreads();

  // softmax over attr dim (i), each thread owns one column n
  if (t < NPIX) {
    float mx = -3.0e38f;
    for (int i = 0; i < NATTR; ++i) mx = fmaxf(mx, qk[i * NPIX + t]);
    float s = 0.f;
    for (int i = 0; i < NATTR; ++i) {
      const float e = __expf(qk[i * NPIX + t] - mx);
      qk[i * NPIX + t] = e;
      s += e;
    }
    const float inv = 1.f / s;
    for (int i = 0; i < NATTR; ++i) qk[i * NPIX + t] *= inv;
  }
  __syncthreads();

  if (t < NATTR) {
    float s = 0.f;
    for (int n = 0; n < NPIX; ++n) s += qk[t * NPIX + n] * g[n];
    out[b * NATTR + t] = s + fc_b[0];
  }
}

// ---------------------------------------------------------------------------
extern "C" void kernel_launch(void* const* d_in, const int* in_sizes, int n_in,
                              void* d_out, int out_size, void* d_ws, size_t ws_size,
                              hipStream_t stream) {
  const float* feature = (const float*)d_in[0];
  const float* Q       = (const float*)d_in[1];
  const float* W1      = (const float*)d_in[2];
  const float* b1      = (const float*)d_in[3];
  const float* W2      = (const float*)d_in[4];
  const float* b2      = (const float*)d_in[5];
  const float* W3      = (const float*)d_in[6];
  const float* fc_w    = (const float*)d_in[7];
  const float* fc_b    = (const float*)d_in[8];
  float* out = (float*)d_out;

  float* ws  = (float*)d_ws;
  float* lq  = ws;                       // 4 * 21504           = 86016 floats
  float* lf1 = lq + (size_t)BATCH * KDIM;        // 4*1024*128  = 524288 floats
  float* lf2 = lf1 + (size_t)BATCH * CCH * NPIX; // 4*1024*128  = 524288 floats
  // total ws use: ~4.54 MB

  ssca_gemv_w3<<<KDIM / 8, 256, 0, stream>>>(Q, W3, lq);
  ssca_lf_gemm<<<(BATCH * 64 * 8) / 8, 256, 0, stream>>>(feature, W1, b1, W2, b2,
                                                         lf1, lf2);
  ssca_attn<<<BATCH, 256, 0, stream>>>(lq, lf1, lf2, fc_w, fc_b, out);
}